// Scattering_85547158602042
// MI455X (gfx1250) — compile-verified
//
#include <hip/hip_runtime.h>
#include <math.h>

// ---------------------------------------------------------------------------
// Scattering transform on MI455X (gfx1250).
// Every FFT2 is two complex GEMMs against a (symmetric) DFT matrix executed
// with V_WMMA_F32_16X16X4_F32.   fft2(X) = ((X*F)^T * F)^T, so the batched
// operand is always the row-major A matrix and F fragments are fetched via
// F's symmetry.  All complex planes use a "pair-split" layout:
//     float4 = { re(2q), re(2q+1), im(2q), im(2q+1) }
// so one b128 load yields both WMMA operand pairs (ar=xy, ai=zw) with no
// register shuffles.  All resolutions are zero-padded to multiples of 16
// (240 / 128 / 64 / 32): no bounds checks, EXEC never touched between WMMAs.
// Forward and conjugated DFT matrices are both materialized (no sign mults).
// ---------------------------------------------------------------------------

typedef float v2f __attribute__((ext_vector_type(2)));
typedef float v8f __attribute__((ext_vector_type(8)));

#define PI_F 3.14159265358979323846f
#define NT 4     // n-tiles (16 cols each) per wave

// Computes Cm = (A * F)^T for square padded size P (P % 16 == 0).
// A: batched [P][P] complex pair-split, F: shared symmetric (maybe conj)
// zero-padded DFT matrix, pair-split.  One wave: 16 rows x 64 cols of A*F.
__global__ __launch_bounds__(128)
void cgemm_rmulF_T(const float4* __restrict__ A,
                   const float4* __restrict__ F,
                   float4* __restrict__ Cm,
                   int P)
{
  int tilesP = P >> 4;
  int tm = blockIdx.y * 4 + threadIdx.y;      // m-tile of this wave (uniform)
  if (tm >= tilesP) return;                   // scalar branch, before any WMMA
  int n0 = blockIdx.x * (NT * 16);

  int Ph = P >> 1;                            // float4 per row
  long plane = (long)P * Ph;                  // float4 per plane
  A  += plane * blockIdx.z;
  Cm += plane * blockIdx.z;

  int lane = threadIdx.x;                     // wave32
  int lan15 = lane & 15;
  int mrow = (tm << 4) + lan15;
  int kho = lane >> 4;                        // lanes 16..31: pair index +1

  // clamp edge tile columns into range (wasted compute, never stored)
  int cbase[NT];
  bool act[NT];
#pragma unroll
  for (int t = 0; t < NT; ++t) {
    int c = n0 + t * 16;
    act[t] = c < P;
    cbase[t] = (c < P ? c : P - 16) + lan15;  // F row index (symmetry)
  }

  v8f cr[NT] = {};
  v8f ci[NT] = {};

  const float4* Arow = A + (long)mrow * Ph + kho;
  const float4* Brow[NT];
#pragma unroll
  for (int t = 0; t < NT; ++t) Brow[t] = F + (long)cbase[t] * Ph + kho;

#pragma unroll 4
  for (int kq = 0; kq < Ph; kq += 2) {        // kq = k0/2, K-step of 4
    // A fragment: {re(ka),re(ka+1),im(ka),im(ka+1)} -> one b128, no shuffle
    float4 av = Arow[kq];
    v2f ar  = { av.x, av.y };
    v2f ai  = { av.z, av.w };
    v2f nai = { -av.z, -av.w };
#pragma unroll
    for (int t = 0; t < NT; ++t) {
      // B fragment via symmetry: F[col][ka..ka+1] -> one b128, no shuffle
      float4 bv = Brow[t][kq];
      v2f br = { bv.x, bv.y };
      v2f bi = { bv.z, bv.w };
      // (ar+i*ai)(br+i*bi): re += ar*br - ai*bi ; im += ar*bi + ai*br
      cr[t] = __builtin_amdgcn_wmma_f32_16x16x4_f32(false, ar,  false, br, (short)0, cr[t], false, false);
      cr[t] = __builtin_amdgcn_wmma_f32_16x16x4_f32(false, nai, false, bi, (short)0, cr[t], false, false);
      ci[t] = __builtin_amdgcn_wmma_f32_16x16x4_f32(false, ar,  false, bi, (short)0, ci[t], false, false);
      ci[t] = __builtin_amdgcn_wmma_f32_16x16x4_f32(false, ai,  false, br, (short)0, ci[t], false, false);
    }
  }

  // D layout: VGPR r -> M=r (lanes 0-15), M=r+8 (lanes 16-31).
  // Transposed store into pair-split: row = col of A*F, pairs along A-rows.
  int rbase = (lane >> 4) << 3;
  int mpair = ((tm << 4) + rbase) >> 1;
#pragma unroll
  for (int t = 0; t < NT; ++t) {
    if (!act[t]) continue;                    // scalar-uniform guard
    float4* cp = Cm + (long)cbase[t] * Ph + mpair;
#pragma unroll
    for (int r = 0; r < 8; r += 2)
      cp[r >> 1] = make_float4(cr[t][r], cr[t][r + 1], ci[t][r], ci[t][r + 1]);
  }
}

// DFT matrix, zero-padded to P x P, pair-split:
// F[k][n] = exp(sign * -2*pi*i*k*n/S) for k,n < S (sign=+1 fwd, -1 conj).
__global__ void dft_kernel(float4* __restrict__ F, int S, int P, float sign) {
  int Ph = P >> 1;
  int id = blockIdx.x * 256 + threadIdx.x;
  if (id >= P * Ph) return;
  int k = id / Ph, n0 = (id % Ph) * 2;
  float c0 = 0.f, s0 = 0.f, c1 = 0.f, s1 = 0.f;
  if (k < S) {
    if (n0 < S) {
      int m = (int)(((long)k * n0) % S);
      float a = -2.0f * PI_F * (float)m / (float)S;
      c0 = cosf(a); s0 = sign * sinf(a);
    }
    if (n0 + 1 < S) {
      int m = (int)(((long)k * (n0 + 1)) % S);
      float a = -2.0f * PI_F * (float)m / (float)S;
      c1 = cosf(a); s1 = sign * sinf(a);
    }
  }
  F[id] = make_float4(c0, c1, s0, s1);
}

// reflect-pad 224 -> 240 (pad 8 each side), pair-split complex (imag = 0)
__global__ void pad_kernel(const float* __restrict__ x, float4* __restrict__ out) {
  long id = (long)blockIdx.x * 256 + threadIdx.x;
  if (id >= 96L * 240 * 120) return;
  int bc  = (int)(id / (240 * 120));
  int rem = (int)(id % (240 * 120));
  int i = rem / 120, j0 = (rem % 120) * 2;
  int si = i - 8;
  si = si < 0 ? -si : (si >= 224 ? 446 - si : si);   // 2*224-2 - si
  int sj0 = j0 - 8, sj1 = j0 - 7;
  sj0 = sj0 < 0 ? -sj0 : (sj0 >= 224 ? 446 - sj0 : sj0);
  sj1 = sj1 < 0 ? -sj1 : (sj1 >= 224 ? 446 - sj1 : sj1);
  const float* row = x + ((long)bc * 224 + si) * 224;
  out[id] = make_float4(row[sj0], row[sj1], 0.f, 0.f);
}

// fused pointwise complex multiply by filter + periodize (k x k block mean).
// U: [96][P][P] pair-split (logical S x S), f: [S][S] interleaved complex
// (raw input filter), out: [96][Pout][Pout] pair-split; pad region zeroed.
// One thread per output column-pair; both columns share the same float4s.
__global__ void mulpz_kernel(const float4* __restrict__ U,
                             const float4* __restrict__ fflt,
                             float4* __restrict__ out,
                             int S, int Pin, int So, int Pout, int k) {
  int PoH = Pout >> 1;
  int t = blockIdx.x * 256 + threadIdx.x;
  if (t >= Pout * PoH) return;
  int b = blockIdx.y;
  int i = t / PoH, jp = t % PoH;               // output row, column-pair
  float4 res = make_float4(0.f, 0.f, 0.f, 0.f);
  if (i < So && jp < (So >> 1)) {
    const float4* Ub = U + (long)b * Pin * (Pin >> 1);
    int j0 = jp * 2;
    float re0 = 0.f, im0 = 0.f, re1 = 0.f, im1 = 0.f;
    for (int a = 0; a < k; ++a) {
      int row = a * So + i;
      for (int c = 0; c < k; ++c) {
        int col0 = c * So + j0;                          // even
        float4 u = Ub[(long)row * (Pin >> 1) + (col0 >> 1)];   // {ur0,ur1,ui0,ui1}
        float4 w = fflt[((long)row * S + col0) >> 1];          // {wr0,wi0,wr1,wi1}
        re0 += u.x * w.x - u.z * w.y;
        im0 += u.x * w.y + u.z * w.x;
        re1 += u.y * w.z - u.w * w.w;
        im1 += u.y * w.w + u.w * w.z;
      }
    }
    float inv = 1.f / (float)(k * k);
    res = make_float4(re0 * inv, re1 * inv, im0 * inv, im1 * inv);
  }
  out[(long)b * Pout * PoH + t] = res;
}

// complex modulus on a pair-split float4: two elements per thread.
__global__ void abs_kernel(const float4* __restrict__ in,
                           float4* __restrict__ out, int PPH) {
  int t = blockIdx.x * 256 + threadIdx.x;
  if (t >= PPH) return;
  long o = (long)blockIdx.y * PPH + t;
  float4 z = in[o];
  out[o] = make_float4(sqrtf(z.x * z.x + z.z * z.z),
                       sqrtf(z.y * z.y + z.w * z.w), 0.f, 0.f);
}

// take real part of padded 32x32 (logical 30x30) inverse transform, unpad to
// 28x28, scatter to output channel block of path p: out[b, p*3+c, i, j]
__global__ void unpad_store_kernel(const float* __restrict__ Vf,
                                   float* __restrict__ out, int p) {
  int t = blockIdx.x * 256 + threadIdx.x;
  if (t >= 784) return;
  int bc = blockIdx.y;                 // 0..95
  int i = t / 28, j = t % 28;
  int row = i + 1, col = j + 1;
  // pair-split: re at ((row*16 + col/2)*4 + (col&1)) within 2048-float plane
  float v = Vf[(long)bc * 2048 + ((row * 16 + (col >> 1)) << 2) + (col & 1)];
  int b = bc / 3, c = bc % 3;
  out[(((long)b * 651 + (long)p * 3 + c) * 28 + i) * 28 + j] = v;
}

extern "C" void kernel_launch(void* const* d_in, const int* in_sizes, int n_in,
                              void* d_out, int out_size, void* d_ws, size_t ws_size,
                              hipStream_t stream) {
  (void)in_sizes; (void)n_in; (void)out_size; (void)ws_size;

  const float* x = (const float*)d_in[0];
  const float4* phiP[3];
  for (int j = 0; j < 3; ++j) phiP[j] = (const float4*)d_in[1 + j];
  // psi flattening: for j1 in 0..2, for t in 0..7, bands r = 0..j1
  const float4* psiP[24][3] = {};
  {
    int idx = 4;
    for (int j1 = 0; j1 < 3; ++j1)
      for (int t = 0; t < 8; ++t) {
        int n1 = j1 * 8 + t;
        for (int r = 0; r <= j1; ++r) psiP[n1][r] = (const float4*)d_in[idx++];
      }
  }

  float* out = (float*)d_out;
  float4* ws = (float4*)d_ws;

  // logical / padded sizes per resolution index
  const int SL[4] = {240, 120, 60, 30};
  const int SP[4] = {240, 128, 64, 32};
  const long FSZ = (240L * 240 + 128 * 128 + 64 * 64 + 32 * 32) / 2; // float4

  // ---- workspace layout (float4 units) ----
  float4* Fm[4];   // forward DFT matrices (pair-split)
  float4* Fc[4];   // conjugated DFT matrices
  {
    float4* q = ws;
    for (int r = 0; r < 4; ++r) { Fm[r] = q; q += (long)SP[r] * SP[r] / 2; }
    for (int r = 0; r < 4; ++r) { Fc[r] = q; q += (long)SP[r] * SP[r] / 2; }
  }
  const long CAP = 96L * 240 * 120;            // float4 per big buffer
  float4* W0  = ws + 2 * FSZ;
  float4* W1  = W0 + CAP;
  float4* W2  = W1 + CAP;
  float4* U0  = W2 + CAP;         // spectrum of padded input
  float4* U1F = U0 + CAP;         // first-order spectrum (reused per n1)

  // ---- helpers ----
  auto rmul = [&](const float4* A, const float4* F, float4* C, int fi) {
    int P = SP[fi];
    dim3 blk(32, 4, 1);
    dim3 grd((unsigned)((P + NT * 16 - 1) / (NT * 16)),
             (unsigned)(((P >> 4) + 3) / 4), 96);
    cgemm_rmulF_T<<<grd, blk, 0, stream>>>(A, F, C, P);
  };
  // Y = opF * X * opF  via two right-multiplies with transposed stores.
  auto fft2 = [&](const float4* X, float4* T, float4* Y, int fi, bool fwd) {
    const float4* F = fwd ? Fm[fi] : Fc[fi];
    rmul(X, F, T, fi);
    rmul(T, F, Y, fi);
  };
  auto mulpz = [&](const float4* U, const float4* f, float4* o, int fiIn, int fiOut) {
    int S = SL[fiIn], Pin = SP[fiIn], So = SL[fiOut], Pout = SP[fiOut];
    int k = 1 << (fiOut - fiIn);
    mulpz_kernel<<<dim3((Pout * (Pout >> 1) + 255) / 256, 96), 256, 0, stream>>>(
        U, f, o, S, Pin, So, Pout, k);
  };
  auto absk = [&](const float4* in, float4* o, int fi) {
    int P = SP[fi];
    abs_kernel<<<dim3((P * (P >> 1) + 255) / 256, 96), 256, 0, stream>>>(
        in, o, P * (P >> 1));
  };
  auto unpad_store = [&](const float4* V, int p) {
    unpad_store_kernel<<<dim3(4, 96), 256, 0, stream>>>((const float*)V, out, p);
  };

  // ---- stage 0: DFT matrices (fwd + conj) + reflect pad ----
  for (int r = 0; r < 4; ++r) {
    int P = SP[r], n = P * (P >> 1);
    dft_kernel<<<(n + 255) / 256, 256, 0, stream>>>(Fm[r], SL[r], P, +1.f);
    dft_kernel<<<(n + 255) / 256, 256, 0, stream>>>(Fc[r], SL[r], P, -1.f);
  }
  pad_kernel<<<(int)((CAP + 255) / 256), 256, 0, stream>>>(x, W0);

  // U0 = fft2(padded input)
  fft2(W0, W1, U0, 0, true);

  // ---- zeroth order ----
  int p = 0;
  mulpz(U0, phiP[0], W1, 0, 3);          // -> 30x30 (padded 32) spectrum
  fft2(W1, W2, W0, 3, false);            // unnormalized inverse
  unpad_store(W0, p); ++p;

  // ---- first + second order ----
  for (int n1 = 0; n1 < 24; ++n1) {
    int j1 = n1 >> 3;

    mulpz(U0, psiP[n1][0], W1, 0, j1);   // U0*psi, periodize -> res j1
    fft2(W1, W2, W0, j1, false);         // spatial U1
    absk(W0, W1, j1);                    // |U1|
    fft2(W1, W2, U1F, j1, true);         // spectrum of |U1|

    // first-order output
    mulpz(U1F, phiP[j1], W1, j1, 3);     // -> 30x30
    fft2(W1, W2, W0, 3, false);
    unpad_store(W0, p); ++p;

    // second order
    for (int n2 = 0; n2 < 24; ++n2) {
      int j2 = n2 >> 3;
      if (j2 <= j1) continue;

      mulpz(U1F, psiP[n2][j1], W1, j1, j2);   // -> res j2
      fft2(W1, W2, W0, j2, false);            // spatial V
      absk(W0, W1, j2);
      fft2(W1, W2, W0, j2, true);             // spectrum of |V|
      mulpz(W0, phiP[j2], W1, j2, 3);         // -> 30x30
      fft2(W1, W2, W0, 3, false);
      unpad_store(W0, p); ++p;
    }
  }
  // p == 217 paths total -> output [32, 651, 28, 28]
}